// DetConLoss_46102178955883
// MI455X (gfx1250) — compile-verified
//
#include <hip/hip_runtime.h>
#include <math.h>

typedef __attribute__((ext_vector_type(2))) float v2f;
typedef __attribute__((ext_vector_type(8))) float v8f;
typedef __attribute__((ext_vector_type(4))) int   v4i;

constexpr int kBN    = 4096;   // B*N
constexpr int kRows  = 8192;   // 2*B*N
constexpr int kE     = 256;    // embedding dim
constexpr int kTile  = 128;    // output macro-tile per block
constexpr int kKC    = 32;     // K chunk staged in LDS
constexpr int kLdsS  = 36;     // padded LDS row stride (float4-aligned, bank-spread)

// Async global->LDS DMA (CDNA5): direct LDS write, tracked by ASYNCcnt.
#if defined(__has_builtin)
#if __has_builtin(__builtin_amdgcn_global_load_async_to_lds_b128)
#define USE_ASYNC_LDS 1
#endif
#endif
#ifndef USE_ASYNC_LDS
#define USE_ASYNC_LDS 0
#endif

#if USE_ASYNC_LDS
typedef __attribute__((address_space(1))) v4i* as1_v4i_ptr;   // global
typedef __attribute__((address_space(3))) v4i* as3_v4i_ptr;   // LDS

__device__ __forceinline__ void async_copy16(const float* gsrc, float* ldst) {
  float* g = const_cast<float*>(gsrc);
  __builtin_amdgcn_global_load_async_to_lds_b128(
      (as1_v4i_ptr)g, (as3_v4i_ptr)ldst, 0, 0);
}
__device__ __forceinline__ void async_wait0() {
#if __has_builtin(__builtin_amdgcn_s_wait_asynccnt)
  __builtin_amdgcn_s_wait_asynccnt(0);
#else
  asm volatile("s_wait_asynccnt 0" ::: "memory");
#endif
}
#endif

// ---------------------------------------------------------------------------
// Kernel 1: L2-normalize over E with temperature scale, write U (8192 x 256)
// Input layout: (B=64, E=256, N=64) -> element (b,e,n) at b*E*64 + e*64 + n
// Output row r = view*4096 + b*64 + n, row-major over e.
// ---------------------------------------------------------------------------
__global__ __launch_bounds__(256) void normalize_kernel(
    const float* __restrict__ view0, const float* __restrict__ view1,
    float* __restrict__ U) {
  const int blk  = blockIdx.x;      // 0..127
  const int view = blk >> 6;
  const int b    = blk & 63;
  const float* src = (view == 0 ? view0 : view1) + (size_t)b * kE * 64;

  __shared__ float partial[256];
  __shared__ float scale[64];

  const int t = threadIdx.x;
  {
    const int n = t & 63, part = t >> 6;
    float ss = 0.f;
    for (int e = part * 64; e < part * 64 + 64; ++e) {
      float x = src[e * 64 + n];
      ss += x * x;
    }
    partial[t] = ss;
  }
  __syncthreads();
  if (t < 64) {
    float ss = partial[t] + partial[t + 64] + partial[t + 128] + partial[t + 192];
    float nrm = sqrtf(ss) + 1e-8f;
    // x / nrm / sqrt(0.1)  ==  x * (1 / (nrm * sqrt(0.1)))
    scale[t] = 1.0f / (nrm * 0.31622776601683794f);
  }
  __syncthreads();

  float* dst = U + (size_t)(view * kBN + b * 64) * kE;
  for (int i = 0; i < 64; ++i) {
    int o = t + 256 * i;            // o = n*256 + e  (n == i here)
    int n = o >> 8, e = o & 255;
    dst[o] = src[e * 64 + n] * scale[n];
  }
}

// ---------------------------------------------------------------------------
// Kernel 2: fused Gram + fixed-shift softmax accumulation.
// G = U U^T (8192x8192). Per element: exp(g - 10) into rowsum (skip diagonal),
// raw logit into pos[] when col == row ^ 4096 (the positive pair).
// All logits bounded by 1/T = 10, so a fixed shift replaces the row-max pass.
// ---------------------------------------------------------------------------
__global__ void zero_kernel(float* __restrict__ p, int n) {
  int i = blockIdx.x * blockDim.x + threadIdx.x;
  if (i < n) p[i] = 0.f;
}

__global__ __launch_bounds__(256) void gram_kernel(
    const float* __restrict__ U, float* __restrict__ rowsum,
    float* __restrict__ pos) {
  __shared__ float As[kTile * kLdsS];
  __shared__ float Bs[kTile * kLdsS];

  const int t    = threadIdx.x;
  const int lane = t & 31;
  const int wave = t >> 5;
  const int tileJ = blockIdx.x * kTile;   // columns
  const int tileI = blockIdx.y * kTile;   // rows

  const v8f vzero = {0.f, 0.f, 0.f, 0.f, 0.f, 0.f, 0.f, 0.f};
  v8f acc[8];
#pragma unroll
  for (int c = 0; c < 8; ++c) acc[c] = vzero;

  const int hi  = lane >> 4;       // 0: lanes 0-15, 1: lanes 16-31
  const int l15 = lane & 15;
  const int arow = wave * 16 + l15;

  for (int kc = 0; kc < kE; kc += kKC) {
    // Stage A(128x32) and B(128x32) tiles into LDS.
#pragma unroll
    for (int i = 0; i < 4; ++i) {
      int s   = t + 256 * i;       // 1024 float4 slots: 128 rows x 8
      int row = s >> 3;
      int c4  = (s & 7) << 2;
#if USE_ASYNC_LDS
      async_copy16(U + (size_t)(tileI + row) * kE + kc + c4,
                   &As[row * kLdsS + c4]);
      async_copy16(U + (size_t)(tileJ + row) * kE + kc + c4,
                   &Bs[row * kLdsS + c4]);
#else
      float4 av = *(const float4*)(U + (size_t)(tileI + row) * kE + kc + c4);
      *(float4*)(&As[row * kLdsS + c4]) = av;
      float4 bv = *(const float4*)(U + (size_t)(tileJ + row) * kE + kc + c4);
      *(float4*)(&Bs[row * kLdsS + c4]) = bv;
#endif
    }
#if USE_ASYNC_LDS
    async_wait0();
#endif
    __syncthreads();

    // 8 K-steps of V_WMMA_F32_16X16X4_F32 per chunk, 8 column tiles each.
    // A 16x4 f32 layout: lanes 0-15 hold K={k0,k0+1}, lanes 16-31 K={k0+2,k0+3}
    // B  4x16 f32 layout mirrors it with N = lane&15.
    for (int k0 = 0; k0 < kKC; k0 += 4) {
      v2f a = *(const v2f*)(&As[arow * kLdsS + k0 + 2 * hi]);
#pragma unroll
      for (int c = 0; c < 8; ++c) {
        v2f bf = *(const v2f*)(&Bs[(c * 16 + l15) * kLdsS + k0 + 2 * hi]);
        acc[c] = __builtin_amdgcn_wmma_f32_16x16x4_f32(
            false, a, false, bf, (short)0, acc[c], false, false);
      }
    }
    __syncthreads();
  }

  // Epilogue: C/D layout: VGPR g holds M = g (lanes 0-15) / M = 8+g (16-31),
  // N = lane & 15. Fixed-shift exp, diag skip, positive capture, row-sum.
  const int rowBase = tileI + wave * 16 + hi * 8;
  float racc[8];
#pragma unroll
  for (int g = 0; g < 8; ++g) racc[g] = 0.f;

  for (int c = 0; c < 8; ++c) {
    const int col = tileJ + c * 16 + l15;
#pragma unroll
    for (int g = 0; g < 8; ++g) {
      const int row = rowBase + g;
      const float v = acc[c][g];
      if (col == (row ^ kBN)) pos[row] = v;          // positive logit
      racc[g] += (row == col) ? 0.f : __expf(v - 10.0f);
    }
  }
#pragma unroll
  for (int g = 0; g < 8; ++g) {
    float v = racc[g];
    v += __shfl_xor(v, 8, 32);    // reduce within each 16-lane half (wave32)
    v += __shfl_xor(v, 4, 32);
    v += __shfl_xor(v, 2, 32);
    v += __shfl_xor(v, 1, 32);
    if (l15 == 0) atomicAdd(&rowsum[rowBase + g], v);
  }
}

// ---------------------------------------------------------------------------
// Kernel 3: loss = mean over rows of (10 - pos[r] + log(rowsum[r]))
// ---------------------------------------------------------------------------
__global__ __launch_bounds__(256) void finalize_kernel(
    const float* __restrict__ rowsum, const float* __restrict__ pos,
    float* __restrict__ out) {
  __shared__ float red[256];
  const int t = threadIdx.x;
  float s = 0.f;
  for (int r = t; r < kRows; r += 256)
    s += 10.0f - pos[r] + logf(rowsum[r]);
  red[t] = s;
  __syncthreads();
  for (int off = 128; off > 0; off >>= 1) {
    if (t < off) red[t] += red[t + off];
    __syncthreads();
  }
  if (t == 0) out[0] = red[0] * (1.0f / (float)kRows);
}

extern "C" void kernel_launch(void* const* d_in, const int* in_sizes, int n_in,
                              void* d_out, int out_size, void* d_ws, size_t ws_size,
                              hipStream_t stream) {
  const float* view0 = (const float*)d_in[0];
  const float* view1 = (const float*)d_in[1];

  float* U      = (float*)d_ws;                 // 8192*256 floats = 8 MB
  float* rowsum = U + (size_t)kRows * kE;       // 8192 floats
  float* pos    = rowsum + kRows;               // 8192 floats

  normalize_kernel<<<128, 256, 0, stream>>>(view0, view1, U);
  zero_kernel<<<(kRows + 255) / 256, 256, 0, stream>>>(rowsum, kRows);
  dim3 grid(kRows / kTile, kRows / kTile);      // 64 x 64 blocks
  gram_kernel<<<grid, 256, 0, stream>>>(U, rowsum, pos);
  finalize_kernel<<<1, 256, 0, stream>>>(rowsum, pos, (float*)d_out);
}